// CRPExpertAggregator_45062796869696
// MI455X (gfx1250) — compile-verified
//
#include <hip/hip_runtime.h>
#include <math.h>

#define B_N 8192
#define F_N 1024
#define E_N 30
#define H_N 256
#define C_N 100
#define G_N 4096

#define MT 32          // rows per expert tile
#define KT 16          // K tile for GEMM1
#define HB (H_N + 4)   // padded stride for B/H tiles: 1040B rows (16B aligned, conflict-free)
#define AP 20          // padded stride for A tile: 80B rows (16B aligned, conflict-free)

typedef __attribute__((ext_vector_type(2))) float v2f;
typedef __attribute__((ext_vector_type(8))) float v8f;

__device__ __forceinline__ float wave_reduce_add(float v) {
#pragma unroll
  for (int m = 16; m > 0; m >>= 1) v += __shfl_xor(v, m, 32);
  return v;
}

__device__ __forceinline__ unsigned lds_addr_u32(const void* p) {
  // flat shared address: low 32 bits == LDS byte offset
  return (unsigned)(size_t)p;
}

__device__ __forceinline__ void async_load_b128(unsigned lds_addr, const void* gaddr) {
  asm volatile("global_load_async_to_lds_b128 %0, %1, off"
               :: "v"(lds_addr), "v"(gaddr) : "memory");
}

__device__ __forceinline__ void wait_async0() {
  asm volatile("s_wait_asynccnt 0x0" ::: "memory");
}

// ---------------- per-expert routing coefficient ----------------
// c_e = 0.5*(1 + cos(g_mem_e, g_new)) * exp(-1.5*max(cnt/5-1,0)) / (||p_e||+eps)
__global__ __launch_bounds__(256) void coef_kernel(
    const float* __restrict__ prototypes, const float* __restrict__ g_new,
    const float* __restrict__ g_mem, const int* __restrict__ class_counts,
    float* __restrict__ c_e) {
  int e = blockIdx.x;
  int tid = threadIdx.x;
  float gdot = 0.f, gmm = 0.f, gnn = 0.f, pp = 0.f;
  for (int i = tid; i < G_N; i += 256) {
    float gm = g_mem[e * G_N + i];
    float gn = g_new[i];
    gdot += gm * gn; gmm += gm * gm; gnn += gn * gn;
  }
  for (int i = tid; i < F_N; i += 256) {
    float p = prototypes[e * F_N + i];
    pp += p * p;
  }
  __shared__ float red[256];
  float vals[4] = {gdot, gmm, gnn, pp};
  float out[4];
  for (int q = 0; q < 4; q++) {
    red[tid] = vals[q];
    __syncthreads();
    for (int s = 128; s > 0; s >>= 1) {
      if (tid < s) red[tid] += red[tid + s];
      __syncthreads();
    }
    out[q] = red[0];
    __syncthreads();
  }
  if (tid == 0) {
    float align = 0.5f * (1.0f + out[0] / ((sqrtf(out[1]) + 1e-8f) * (sqrtf(out[2]) + 1e-8f)));
    float overload = fmaxf((float)class_counts[e] / 5.0f - 1.0f, 0.0f);
    float capacity = expf(-1.5f * overload);
    c_e[e] = align * capacity / (sqrtf(out[3]) + 1e-8f);
  }
}

__global__ void init_kernel(int* counts) {
  if (threadIdx.x < E_N) counts[threadIdx.x] = 0;
}

// ---------------- routing: one wave32 per sample ----------------
__global__ __launch_bounds__(256) void route_kernel(
    const float* __restrict__ x, const float* __restrict__ prototypes,
    const float* __restrict__ c_e, int* __restrict__ assign, int* __restrict__ counts) {
  __shared__ float sC[E_N];
  int tid = threadIdx.x;
  if (tid < E_N) sC[tid] = c_e[tid];
  __syncthreads();
  int w = tid >> 5, lane = tid & 31;
  int b = blockIdx.x * 8 + w;
  float acc[E_N];
#pragma unroll
  for (int e = 0; e < E_N; e++) acc[e] = 0.f;
  const float* xr = x + (size_t)b * F_N;
  for (int k = 0; k < F_N / 32; k++) {
    float xv = xr[k * 32 + lane];
#pragma unroll
    for (int e = 0; e < E_N; e++)
      acc[e] += xv * prototypes[e * F_N + k * 32 + lane];
  }
  float best = -1e30f;
  int bi = 0;
#pragma unroll
  for (int e = 0; e < E_N; e++) {
    float v = wave_reduce_add(acc[e]) * sC[e];
    if (v > best) { best = v; bi = e; }   // strict > : first max, matches jnp.argmax
  }
  if (lane == 0) {
    assign[b] = bi;
    atomicAdd(&counts[bi], 1);
  }
}

__global__ void scan_kernel(const int* __restrict__ counts, int* __restrict__ offsets,
                            int* __restrict__ cursor) {
  if (threadIdx.x == 0) {
    int acc = 0;
    for (int e = 0; e < E_N; e++) {
      offsets[e] = acc;
      cursor[e] = acc;
      acc += counts[e];
    }
    offsets[E_N] = acc;
  }
}

__global__ __launch_bounds__(256) void scatter_kernel(const int* __restrict__ assign,
                                                      int* __restrict__ cursor,
                                                      int* __restrict__ perm) {
  int b = blockIdx.x * 256 + threadIdx.x;
  int e = assign[b];
  int pos = atomicAdd(&cursor[e], 1);
  perm[pos] = b;
}

// ---------------- grouped expert MLP ----------------
// gather -> GEMM1 (async double-buffered LDS + f32 WMMA) -> LN -> GELU -> GEMM2 -> scatter
__global__ __launch_bounds__(256) void expert_mlp_kernel(
    const float* __restrict__ x, const float* __restrict__ W1,
    const float* __restrict__ b1, const float* __restrict__ gamma,
    const float* __restrict__ beta, const float* __restrict__ W2,
    const float* __restrict__ b2, const int* __restrict__ perm,
    const int* __restrict__ offsets, const int* __restrict__ counts,
    float* __restrict__ out) {
  int e = blockIdx.x;
  int tile = blockIdx.y;
  int cnt = counts[e];
  if (tile * MT >= cnt) return;
  int base = offsets[e];

  // As: double-buffered gathered x tile (async target, padded stride AP)
  // BH: double-buffered W1 k-tile during GEMM1 (2 x KT x HB), then aliased as
  //     the 32 x HB activation tile afterwards (2*KT == MT, sizes identical).
  __shared__ __align__(16) float As[2][MT][AP];
  __shared__ __align__(16) float BH[MT * HB];
  __shared__ int rows[MT];
  __shared__ int rowsafe[MT];

  int tid = threadIdx.x;
  int w = tid >> 5, lane = tid & 31;
  if (tid < MT) {
    int s = tile * MT + tid;
    int rb = (s < cnt) ? perm[base + s] : -1;
    rows[tid] = rb;
    rowsafe[tid] = (rb >= 0) ? rb : perm[base];  // padding slots -> any valid row
  }
  __syncthreads();

  int mstrip = w >> 2;        // 0..1 : which 16-row strip
  int ng = (w & 3) * 64;      // 64-column group of H

  const int NK = F_N / KT;    // 64 k-tiles

  // ---- async tile loader: W1 k-tile (16x256) + gathered x tile (32x16) ----
  auto load_tile = [&](int kt, int buf) {
    int k0 = kt * KT;
    // B tile: 4096 floats = 1024 16B-chunks; 4 perfectly-coalesced b128 per thread
    unsigned lbase = lds_addr_u32(BH) + (unsigned)(buf * KT * HB * 4);
    const float* gbase = W1 + ((size_t)e * F_N + k0) * H_N;
#pragma unroll
    for (int j = 0; j < 4; j++) {
      int c = j * 256 + tid;            // chunk id
      int row = c >> 6;                 // 64 chunks per 256-float row
      int col = (c & 63) * 4;
      async_load_b128(lbase + (unsigned)((row * HB + col) * 4),
                      gbase + (size_t)row * H_N + col);
    }
    // A tile: 32 rows x 16 floats = 128 chunks; threads 0..127
    if (tid < 128) {
      int row = tid >> 2;
      int col = (tid & 3) * 4;
      unsigned laddr = lds_addr_u32(&As[0][0][0]) +
                       (unsigned)(((buf * MT + row) * AP + col) * 4);
      async_load_b128(laddr, x + (size_t)rowsafe[row] * F_N + k0 + col);
    }
  };

  v8f acc[4] = {};
  load_tile(0, 0);
  wait_async0();
  __syncthreads();

  for (int kt = 0; kt < NK; ++kt) {
    int buf = kt & 1;
    if (kt + 1 < NK) load_tile(kt + 1, buf ^ 1);   // prefetch next tile (overlapped)

    int ksel = (lane >= 16) ? 2 : 0;   // f32 WMMA layout: lanes 16-31 hold K+2/K+3
    int mm = mstrip * 16 + (lane & 15);
    int nn = ng + (lane & 15);
    const float* Bbuf = BH + buf * KT * HB;
#pragma unroll
    for (int kk = 0; kk < KT; kk += 4) {
      v2f a;
      a.x = As[buf][mm][kk + ksel];
      a.y = As[buf][mm][kk + ksel + 1];
#pragma unroll
      for (int t = 0; t < 4; t++) {
        v2f bf;
        bf.x = Bbuf[(kk + ksel) * HB + nn + t * 16];
        bf.y = Bbuf[(kk + ksel + 1) * HB + nn + t * 16];
        acc[t] = __builtin_amdgcn_wmma_f32_16x16x4_f32(
            false, a, false, bf, (short)0, acc[t], false, false);
      }
    }
    wait_async0();     // next-tile loads had the whole compute window to land
    __syncthreads();
  }

  // ---- h (+bias) into LDS (BH now = activation tile Hs[m][HB]) ----
#pragma unroll
  for (int t = 0; t < 4; t++) {
#pragma unroll
    for (int r = 0; r < 8; r++) {
      int m = mstrip * 16 + ((lane < 16) ? r : r + 8);
      int n = ng + t * 16 + (lane & 15);
      BH[m * HB + n] = acc[t][r] + b1[e * H_N + n];
    }
  }
  __syncthreads();

  // ---- LayerNorm + exact-erf GELU, one wave handles 4 rows ----
  for (int rr = 0; rr < 4; rr++) {
    int r = w * 4 + rr;
    float s = 0.f;
#pragma unroll
    for (int j = 0; j < H_N / 32; j++) s += BH[r * HB + lane + j * 32];
    s = wave_reduce_add(s);
    float mean = s * (1.0f / H_N);
    float v = 0.f;
#pragma unroll
    for (int j = 0; j < H_N / 32; j++) {
      float d = BH[r * HB + lane + j * 32] - mean;
      v += d * d;
    }
    v = wave_reduce_add(v);
    float rstd = rsqrtf(v * (1.0f / H_N) + 1e-5f);
#pragma unroll
    for (int j = 0; j < H_N / 32; j++) {
      int n = lane + j * 32;
      float val = (BH[r * HB + n] - mean) * rstd * gamma[e * H_N + n] + beta[e * H_N + n];
      val = 0.5f * val * (1.0f + erff(val * 0.70710678118654752f));
      BH[r * HB + n] = val;
    }
  }
  __syncthreads();

  // ---- GEMM2: [32 x 256] @ [256 x 100]; 2 m-strips x 7 n-tiles = 14 units / 8 waves ----
  for (int u = w; u < 14; u += 8) {
    int ms = u / 7, nt = u % 7;
    v8f c = {};
    int ksel = (lane >= 16) ? 2 : 0;
    int mm = ms * 16 + (lane & 15);
    int nn = nt * 16 + (lane & 15);
    bool nok = nn < C_N;
#pragma unroll 4
    for (int k0 = 0; k0 < H_N; k0 += 4) {
      v2f a;
      a.x = BH[mm * HB + k0 + ksel];
      a.y = BH[mm * HB + k0 + ksel + 1];
      v2f bf;
      bf.x = nok ? W2[((size_t)e * H_N + k0 + ksel) * C_N + nn] : 0.f;
      bf.y = nok ? W2[((size_t)e * H_N + k0 + ksel + 1) * C_N + nn] : 0.f;
      c = __builtin_amdgcn_wmma_f32_16x16x4_f32(
          false, a, false, bf, (short)0, c, false, false);
    }
    if (nok) {
      float bb = b2[e * C_N + nn];
#pragma unroll
      for (int r = 0; r < 8; r++) {
        int m = ms * 16 + ((lane < 16) ? r : r + 8);
        int rb = rows[m];
        if (rb >= 0) out[(size_t)rb * C_N + nn] = c[r] + bb;
      }
    }
  }
}

extern "C" void kernel_launch(void* const* d_in, const int* in_sizes, int n_in,
                              void* d_out, int out_size, void* d_ws, size_t ws_size,
                              hipStream_t stream) {
  const float* x           = (const float*)d_in[0];
  const float* prototypes  = (const float*)d_in[1];
  const float* g_new       = (const float*)d_in[2];
  const float* g_mem       = (const float*)d_in[3];
  const int*   class_counts= (const int*)d_in[4];
  const float* W1          = (const float*)d_in[5];
  const float* b1          = (const float*)d_in[6];
  const float* gamma       = (const float*)d_in[7];
  const float* beta        = (const float*)d_in[8];
  const float* W2          = (const float*)d_in[9];
  const float* b2          = (const float*)d_in[10];
  float* out = (float*)d_out;

  // workspace carve
  float* c_e    = (float*)d_ws;           // 32
  int*   assign = (int*)(c_e + 32);       // B_N
  int*   counts = assign + B_N;           // 32
  int*   offsets= counts + 32;            // 32
  int*   cursor = offsets + 32;           // 32
  int*   perm   = cursor + 32;            // B_N

  coef_kernel<<<E_N, 256, 0, stream>>>(prototypes, g_new, g_mem, class_counts, c_e);
  init_kernel<<<1, 64, 0, stream>>>(counts);
  route_kernel<<<B_N / 8, 256, 0, stream>>>(x, prototypes, c_e, assign, counts);
  scan_kernel<<<1, 32, 0, stream>>>(counts, offsets, cursor);
  scatter_kernel<<<B_N / 256, 256, 0, stream>>>(assign, cursor, perm);
  dim3 g(E_N, (B_N + MT - 1) / MT);
  expert_mlp_kernel<<<g, 256, 0, stream>>>(x, W1, b1, gamma, beta, W2, b2,
                                           perm, offsets, counts, out);
}